// ButterworthIIR_39247411151252
// MI455X (gfx1250) — compile-verified
//
#include <hip/hip_runtime.h>

typedef __attribute__((ext_vector_type(2))) float v2f;
typedef __attribute__((ext_vector_type(8))) float v8f;

#define T_LEN   160000
#define NBATCH  128
#define CHUNK   800
#define NCHUNK  200            // T_LEN / CHUNK

// workspace layout (float offsets)
#define AL_OFF   0                                   // 16 floats: A^CHUNK row-major
#define HHOM_OFF 16                                  // CHUNK*4 floats: Hhom[k] = e0^T A^k
#define ZIN_OFF  (16 + CHUNK * 4)                    // NCHUNK*NBATCH*4: chunk-entry states
#define DST_OFF  (ZIN_OFF + NCHUNK * NBATCH * 4)     // NCHUNK*NBATCH*4: zero-state end states

// ---------------------------------------------------------------------------
// Kernel 0: build A (4x4 companion form of DF2T), Hhom table, and A^CHUNK.
// Single lane; ~50k serial FLOPs, negligible.
// ---------------------------------------------------------------------------
__global__ void k0_setup(const float* __restrict__ bc, const float* __restrict__ ac,
                         float* __restrict__ ws) {
  if (threadIdx.x != 0 || blockIdx.x != 0) return;
  float inv = 1.0f / ac[0];
  float an1 = ac[1] * inv, an2 = ac[2] * inv, an3 = ac[3] * inv, an4 = ac[4] * inv;
  // A = upshift - avec * e0^T
  float A[16];
  for (int i = 0; i < 16; ++i) A[i] = 0.0f;
  A[0]  = -an1; A[1]  = 1.0f;
  A[4]  = -an2; A[6]  = 1.0f;
  A[8]  = -an3; A[11] = 1.0f;
  A[12] = -an4;
  // Hhom[k] = e0^T A^k  (row vector iteration)
  float r0 = 1.f, r1 = 0.f, r2 = 0.f, r3 = 0.f;
  for (int k = 0; k < CHUNK; ++k) {
    float* h = ws + HHOM_OFF + k * 4;
    h[0] = r0; h[1] = r1; h[2] = r2; h[3] = r3;
    float n0 = r0 * A[0] + r1 * A[4] + r2 * A[8]  + r3 * A[12];
    float n1 = r0 * A[1] + r1 * A[5] + r2 * A[9]  + r3 * A[13];
    float n2 = r0 * A[2] + r1 * A[6] + r2 * A[10] + r3 * A[14];
    float n3 = r0 * A[3] + r1 * A[7] + r2 * A[11] + r3 * A[15];
    r0 = n0; r1 = n1; r2 = n2; r3 = n3;
  }
  // AL = A^CHUNK by repeated multiply
  float M[16];
  for (int i = 0; i < 16; ++i) M[i] = (i % 5 == 0) ? 1.0f : 0.0f;
  for (int k = 0; k < CHUNK; ++k) {
    float NM[16];
    for (int i = 0; i < 4; ++i)
      for (int j = 0; j < 4; ++j) {
        float s = 0.0f;
        for (int t = 0; t < 4; ++t) s = fmaf(A[i * 4 + t], M[t * 4 + j], s);
        NM[i * 4 + j] = s;
      }
    for (int i = 0; i < 16; ++i) M[i] = NM[i];
  }
  for (int i = 0; i < 16; ++i) ws[AL_OFF + i] = M[i];
}

// ---------------------------------------------------------------------------
// Kernel 1: per (chunk,batch) zero-state recurrence; writes y_zs and end-state D.
// ---------------------------------------------------------------------------
__global__ void k1_zerostate(const float* __restrict__ x,
                             const float* __restrict__ bc, const float* __restrict__ ac,
                             float* __restrict__ y, float* __restrict__ ws) {
  int tid = blockIdx.x * blockDim.x + threadIdx.x;
  if (tid >= NCHUNK * NBATCH) return;
  int c = tid % NCHUNK;
  int b = tid / NCHUNK;
  float inv = 1.0f / ac[0];
  float b0 = bc[0] * inv, b1 = bc[1] * inv, b2 = bc[2] * inv, b3 = bc[3] * inv, b4 = bc[4] * inv;
  float a1 = ac[1] * inv, a2 = ac[2] * inv, a3 = ac[3] * inv, a4 = ac[4] * inv;
  float z0 = 0.f, z1 = 0.f, z2 = 0.f, z3 = 0.f;
  size_t base = (size_t)b * T_LEN + (size_t)c * CHUNK;
  const float4* xp = reinterpret_cast<const float4*>(x + base);
  float4* yp = reinterpret_cast<float4*>(y + base);
#define IIR_STEP(XV, YV)                              \
  {                                                   \
    float yy = fmaf(b0, (XV), z0);                    \
    (YV) = yy;                                        \
    z0 = fmaf(-a1, yy, fmaf(b1, (XV), z1));           \
    z1 = fmaf(-a2, yy, fmaf(b2, (XV), z2));           \
    z2 = fmaf(-a3, yy, fmaf(b3, (XV), z3));           \
    z3 = fmaf(-a4, yy, b4 * (XV));                    \
  }
  for (int i = 0; i < CHUNK / 4; ++i) {
    float4 xv = xp[i];
    float4 yv;
    IIR_STEP(xv.x, yv.x);
    IIR_STEP(xv.y, yv.y);
    IIR_STEP(xv.z, yv.z);
    IIR_STEP(xv.w, yv.w);
    yp[i] = yv;
  }
#undef IIR_STEP
  float* d = ws + DST_OFF + (size_t)(c * NBATCH + b) * 4;
  d[0] = z0; d[1] = z1; d[2] = z2; d[3] = z3;
}

// ---------------------------------------------------------------------------
// Kernel 2: sequential scan over chunks (one lane per batch):
//   Zin[c] = z;  z = A^CHUNK * z + D[c]
// ---------------------------------------------------------------------------
__global__ void k2_scan(float* __restrict__ ws) {
  int b = threadIdx.x;
  if (b >= NBATCH) return;
  float M[16];
  for (int i = 0; i < 16; ++i) M[i] = ws[AL_OFF + i];
  float z0 = 0.f, z1 = 0.f, z2 = 0.f, z3 = 0.f;
  for (int c = 0; c < NCHUNK; ++c) {
    size_t o = (size_t)(c * NBATCH + b) * 4;
    float* zo = ws + ZIN_OFF + o;
    zo[0] = z0; zo[1] = z1; zo[2] = z2; zo[3] = z3;
    const float* d = ws + DST_OFF + o;
    float n0 = fmaf(M[0],  z0, fmaf(M[1],  z1, fmaf(M[2],  z2, fmaf(M[3],  z3, d[0]))));
    float n1 = fmaf(M[4],  z0, fmaf(M[5],  z1, fmaf(M[6],  z2, fmaf(M[7],  z3, d[1]))));
    float n2 = fmaf(M[8],  z0, fmaf(M[9],  z1, fmaf(M[10], z2, fmaf(M[11], z3, d[2]))));
    float n3 = fmaf(M[12], z0, fmaf(M[13], z1, fmaf(M[14], z2, fmaf(M[15], z3, d[3]))));
    z0 = n0; z1 = n1; z2 = n2; z3 = n3;
  }
}

// ---------------------------------------------------------------------------
// Kernel 3: homogeneous correction via V_WMMA_F32_16X16X4_F32.
// One wave per tile (chunk c, batch-group bg, offset-group og):
//   corr[16x16] = Hhom[og*16 .. og*16+15][0..3] (16x4) @ Zin_c[4 x 16 batches]
//   y[b][t] += corr
// ---------------------------------------------------------------------------
__global__ void k3_corr(float* __restrict__ y, const float* __restrict__ ws) {
  const int lane = threadIdx.x & 31;
  const int wave = (blockIdx.x * (int)blockDim.x + threadIdx.x) >> 5;
  const int OG = CHUNK / 16;   // 50
  const int BG = NBATCH / 16;  // 8
  int og = wave % OG;
  int bg = (wave / OG) % BG;
  int c  = wave / (OG * BG);
  if (c >= NCHUNK) return;  // wave-uniform; grid is exact

  const int half  = lane >> 4;   // 0: K0/K1 (or M rows 0-7 of C), 1: K2/K3 (rows 8-15)
  const int idx16 = lane & 15;

  // A fragment: 16x4 f32, lanes 0-15 hold {K0,K1}, lanes 16-31 hold {K2,K3}
  const float* hh = ws + HHOM_OFF + (size_t)(og * 16 + idx16) * 4 + 2 * half;
  v2f afrag;
  afrag.x = hh[0];
  afrag.y = hh[1];

  // B fragment: 4x16 f32, lane holds column n = idx16, rows {2*half, 2*half+1}
  const float* zp = ws + ZIN_OFF + (size_t)(c * NBATCH + bg * 16 + idx16) * 4 + 2 * half;
  v2f bfrag;
  bfrag.x = zp[0];
  bfrag.y = zp[1];

  v8f acc = {};
  acc = __builtin_amdgcn_wmma_f32_16x16x4_f32(
      /*neg_a=*/false, afrag, /*neg_b=*/false, bfrag,
      /*c_mod=*/(short)0, acc, /*reuse_a=*/false, /*reuse_b=*/false);

  // C/D layout: lane n=idx16, VGPR r -> M = half*8 + r  => 8 consecutive t per lane
  int batch = bg * 16 + idx16;
  size_t t0 = (size_t)c * CHUNK + og * 16 + half * 8;
  float* p = y + (size_t)batch * T_LEN + t0;
  float4 v0 = *reinterpret_cast<float4*>(p);
  float4 v1 = *reinterpret_cast<float4*>(p + 4);
  v0.x += acc[0]; v0.y += acc[1]; v0.z += acc[2]; v0.w += acc[3];
  v1.x += acc[4]; v1.y += acc[5]; v1.z += acc[6]; v1.w += acc[7];
  *reinterpret_cast<float4*>(p) = v0;
  *reinterpret_cast<float4*>(p + 4) = v1;
}

// ---------------------------------------------------------------------------
extern "C" void kernel_launch(void* const* d_in, const int* in_sizes, int n_in,
                              void* d_out, int out_size, void* d_ws, size_t ws_size,
                              hipStream_t stream) {
  (void)in_sizes; (void)n_in; (void)out_size; (void)ws_size;
  const float* x  = (const float*)d_in[0];
  const float* bc = (const float*)d_in[1];
  const float* ac = (const float*)d_in[2];
  float* y  = (float*)d_out;
  float* ws = (float*)d_ws;

  k0_setup<<<1, 32, 0, stream>>>(bc, ac, ws);

  int n1 = NCHUNK * NBATCH;  // 25600 threads
  k1_zerostate<<<(n1 + 255) / 256, 256, 0, stream>>>(x, bc, ac, y, ws);

  k2_scan<<<1, 128, 0, stream>>>(ws);

  int tiles = NCHUNK * (NBATCH / 16) * (CHUNK / 16);  // 80000 waves
  int blocks = tiles / 8;                             // 8 waves (256 threads) per block
  k3_corr<<<blocks, 256, 0, stream>>>(y, ws);
}